// SpecSteerSampler_69724499084028
// MI455X (gfx1250) — compile-verified
//
#include <hip/hip_runtime.h>
#include <math.h>
#include <stdint.h>

// ---------------- problem constants (match reference) ----------------
#define B_   64
#define K_   8
#define V_   50257
#define T_   20
#define GAMMA_ 0.6f
#define EPS_   1e-10f
#define ALPHA_ 2.0f
#define BETA_  1.5f
#define LAM_   2.0f
#define ETA_   10.0f
#define PLACEHOLDER_ (-1)

// ---------------- helpers ----------------
__device__ __forceinline__ float nan_to(float x, float v) { return (x != x) ? v : x; }

// combine two online-logsumexp states (m = running max, s = sum exp(x-m))
__device__ __forceinline__ void lse_combine(float& m, float& s, float mo, float so) {
    float M = fmaxf(m, mo);
    if (M == -INFINITY) { m = M; s = 0.0f; return; }   // both empty
    s = s * expf(m - M) + so * expf(mo - M);
    m = M;
}

__device__ __forceinline__ float warp_max(float v) {
    #pragma unroll
    for (int o = 16; o > 0; o >>= 1) v = fmaxf(v, __shfl_xor(v, o, 32));
    return v;
}
__device__ __forceinline__ float warp_sum(float v) {
    #pragma unroll
    for (int o = 16; o > 0; o >>= 1) v += __shfl_xor(v, o, 32);
    return v;
}

// block-wide reductions (blockDim.x multiple of 32, <=1024). smem: >=32 floats.
__device__ float block_max(float v, float* smem) {
    int lane = threadIdx.x & 31, w = threadIdx.x >> 5;
    int nw = blockDim.x >> 5;
    v = warp_max(v);
    if (lane == 0) smem[w] = v;
    __syncthreads();
    if (w == 0) {
        float x = (lane < nw) ? smem[lane] : -INFINITY;
        x = warp_max(x);
        if (lane == 0) smem[0] = x;
    }
    __syncthreads();
    float r = smem[0];
    __syncthreads();
    return r;
}
__device__ float block_sum(float v, float* smem) {
    int lane = threadIdx.x & 31, w = threadIdx.x >> 5;
    int nw = blockDim.x >> 5;
    v = warp_sum(v);
    if (lane == 0) smem[w] = v;
    __syncthreads();
    if (w == 0) {
        float x = (lane < nw) ? smem[lane] : 0.0f;
        x = warp_sum(x);
        if (lane == 0) smem[0] = x;
    }
    __syncthreads();
    float r = smem[0];
    __syncthreads();
    return r;
}

// ---------------- kernel 1: per-row softmax prob at draft token ----------------
// One block per row r in [0, B*K). Online logsumexp over V elements of the
// target and base rows. The 206 MB stream goes through the CDNA5 async
// global->LDS DMA path (ASYNCcnt), double-buffered so chunk c+1 is in flight
// while chunk c is consumed by the VALU softmax accumulation.
#define TILE_A 2048   // floats per tensor per chunk (8 KB)
#define THR_A  256
#define NCH_A  ((V_ + TILE_A - 1) / TILE_A)   // 25 chunks
// per wave per chunk: (TILE_A/THR_A) slots x 2 tensors = 16 async instructions
#define ASY_PER_CHUNK 16

__global__ __launch_bounds__(THR_A) void probs_kernel(
    const float* __restrict__ target, const float* __restrict__ base,
    const int* __restrict__ draft, float* __restrict__ p_llm,
    float* __restrict__ p_base) {
    const int row = blockIdx.x;
    const float* trow = target + (size_t)row * V_;
    const float* brow = base   + (size_t)row * V_;

    __shared__ float tbuf[2][TILE_A];
    __shared__ float bbuf[2][TILE_A];
    __shared__ float wm_t[8], ws_t[8], wm_b[8], ws_b[8];

    const int tid = threadIdx.x;
    // low 32 bits of a generic pointer to LDS == byte offset within the
    // workgroup's LDS allocation (aperture layout; see ISA 10.2)
    const unsigned tb = (unsigned)(uintptr_t)(&tbuf[0][0]);
    const unsigned bb = (unsigned)(uintptr_t)(&bbuf[0][0]);

    // issue one chunk's async loads into LDS buffer `buf`.
    // Tail addresses are clamped (not skipped) so every chunk issues exactly
    // ASY_PER_CHUNK async instructions per wave -> exact s_wait_asynccnt math.
    auto issue = [&](int c, int buf) {
        const int cbase = c * TILE_A;
        #pragma unroll
        for (int j = 0; j < TILE_A / THR_A; ++j) {
            const int slot = tid + j * THR_A;
            int idx = cbase + slot;
            idx = (idx < V_) ? idx : (V_ - 1);
            const unsigned goff = (unsigned)idx * 4u;
            const unsigned ldst = tb + (unsigned)(buf * TILE_A + slot) * 4u;
            const unsigned ldsb = bb + (unsigned)(buf * TILE_A + slot) * 4u;
            asm volatile("global_load_async_to_lds_b32 %0, %1, %2"
                         :: "v"(ldst), "v"(goff), "s"(trow) : "memory");
            asm volatile("global_load_async_to_lds_b32 %0, %1, %2"
                         :: "v"(ldsb), "v"(goff), "s"(brow) : "memory");
        }
    };

    float m_t = -INFINITY, s_t = 0.0f;
    float m_b = -INFINITY, s_b = 0.0f;

    issue(0, 0);
    for (int c = 0; c < NCH_A; ++c) {
        const int buf = c & 1;
        if (c + 1 < NCH_A) {
            issue(c + 1, (c + 1) & 1);
            // allow next chunk's loads to remain outstanding
            asm volatile("s_wait_asynccnt %0" :: "i"(ASY_PER_CHUNK) : "memory");
        } else {
            asm volatile("s_wait_asynccnt 0" ::: "memory");
        }
        __syncthreads();
        // consume the tile: online logsumexp (1 expf/elt in the common path)
        #pragma unroll
        for (int j = 0; j < TILE_A / THR_A; ++j) {
            const int slot = tid + j * THR_A;
            const int idx  = c * TILE_A + slot;
            if (idx < V_) {
                float x = tbuf[buf][slot];
                if (x <= m_t) { s_t += expf(x - m_t); }
                else          { s_t = s_t * expf(m_t - x) + 1.0f; m_t = x; }
                x = bbuf[buf][slot];
                if (x <= m_b) { s_b += expf(x - m_b); }
                else          { s_b = s_b * expf(m_b - x) + 1.0f; m_b = x; }
            }
        }
        __syncthreads();   // all reads done before this buffer is re-filled
    }

    // combine (m,s) pairs across the wave, then across waves
    #pragma unroll
    for (int o = 16; o > 0; o >>= 1) {
        float mo = __shfl_xor(m_t, o, 32), so = __shfl_xor(s_t, o, 32);
        lse_combine(m_t, s_t, mo, so);
        mo = __shfl_xor(m_b, o, 32); so = __shfl_xor(s_b, o, 32);
        lse_combine(m_b, s_b, mo, so);
    }
    const int lane = tid & 31, w = tid >> 5;
    if (lane == 0) { wm_t[w] = m_t; ws_t[w] = s_t; wm_b[w] = m_b; ws_b[w] = s_b; }
    __syncthreads();
    if (tid == 0) {
        float Mt = wm_t[0], St = ws_t[0], Mb = wm_b[0], Sb = ws_b[0];
        #pragma unroll
        for (int i = 1; i < THR_A / 32; ++i) {
            lse_combine(Mt, St, wm_t[i], ws_t[i]);
            lse_combine(Mb, Sb, wm_b[i], ws_b[i]);
        }
        const float lse_t = Mt + logf(St);
        const float lse_b = Mb + logf(Sb);
        const int id = draft[row];
        p_llm[row]  = expf(trow[id] - lse_t);
        p_base[row] = expf(brow[id] - lse_b);
    }
}

// ---------------- kernel 2: acceptance logic ----------------
__global__ void accept_kernel(const float* __restrict__ p_llm,
                              const float* __restrict__ p_base,
                              const int* __restrict__ num_draft,
                              int* __restrict__ cnt_out,   // -> d_out cnt block
                              int* __restrict__ rp,
                              int* __restrict__ rejected) {
    const int b = threadIdx.x;
    if (b >= B_) return;
    const int n = num_draft[b];
    int cnt = 0;
    bool ok = true;
    #pragma unroll
    for (int k = 0; k < K_; ++k) {
        const float pl = p_llm[b * K_ + k];
        const float pb = p_base[b * K_ + k];
        const bool a = (pl > GAMMA_ * (pb + EPS_)) && (k < n);
        ok = ok && a;
        cnt += ok ? 1 : 0;
    }
    cnt_out[b]  = cnt;
    rp[b]       = (cnt < K_ - 1) ? cnt : (K_ - 1);
    rejected[b] = (cnt < n) ? 1 : 0;
}

// ---------------- kernel 3: costeer fused distribution ----------------
// One block (512 threads = 16 wave32) per request b. prev/log_p and q_sum
// live entirely in VGPRs (99 elements/thread, ~200 regs; 4 waves/SIMD so the
// per-wave VGPR budget is maximal). Per iteration the only memory traffic is
// re-reading the three L2-resident logit rows; reductions are shuffle+LDS.
#define THR_C   512
#define ITEMS_C ((V_ + THR_C - 1) / THR_C)   // 99

__global__ __launch_bounds__(THR_C, 1) void costeer_kernel(
    const float* __restrict__ target, const float* __restrict__ base,
    const float* __restrict__ steer, const int* __restrict__ rp,
    float* __restrict__ fused, int* __restrict__ recovered) {
    const int b = blockIdx.x;
    const int r = rp[b];
    const float* tr = target + ((size_t)b * K_ + r) * V_;
    const float* br = base   + ((size_t)b * K_ + r) * V_;
    const float* sr = steer  + ((size_t)b * K_ + r) * V_;
    float* outrow = fused + (size_t)b * V_;

    __shared__ float sred[32];
    __shared__ int   sredi[32];
    const int tid = threadIdx.x;

    // ---- phase 0: lse of nan_to(x,-100) for the three rows ----
    float mt = -INFINITY, mb = -INFINITY, ms = -INFINITY;
    for (int v = tid; v < V_; v += THR_C) {
        mt = fmaxf(mt, nan_to(tr[v], -100.0f));
        mb = fmaxf(mb, nan_to(br[v], -100.0f));
        ms = fmaxf(ms, nan_to(sr[v], -100.0f));
    }
    mt = block_max(mt, sred);
    mb = block_max(mb, sred);
    ms = block_max(ms, sred);
    float st = 0.0f, sb = 0.0f, ss = 0.0f;
    for (int v = tid; v < V_; v += THR_C) {
        st += expf(nan_to(tr[v], -100.0f) - mt);
        sb += expf(nan_to(br[v], -100.0f) - mb);
        ss += expf(nan_to(sr[v], -100.0f) - ms);
    }
    st = block_sum(st, sred);
    sb = block_sum(sb, sred);
    ss = block_sum(ss, sred);
    const float lse_t = mt + logf(st);
    const float lse_b = mb + logf(sb);
    const float lse_s = ms + logf(ss);

    // ---- register-resident state: prev (== log_p == lp) and q_sum ----
    float prev[ITEMS_C];
    float qsr[ITEMS_C];
    #pragma unroll
    for (int i = 0; i < ITEMS_C; ++i) {
        const int v = tid + i * THR_C;
        prev[i] = (v < V_) ? (nan_to(tr[v], -100.0f) - lse_t) : -INFINITY;
        qsr[i]  = 0.0f;
    }

    // ---- T iterations, zero global traffic except logit-row reloads ----
    for (int t = 1; t <= T_; ++t) {
        const float tf = (float)t;
        const float denom = tf * LAM_ + 1.0f / ETA_;
        float mraw = -INFINITY;
        #pragma unroll
        for (int i = 0; i < ITEMS_C; ++i) {
            const int v = tid + i * THR_C;
            if (v < V_) {
                const float llm = nan_to(tr[v], -100.0f) - lse_t;
                float d = (nan_to(sr[v], -100.0f) - lse_s)
                        - (nan_to(br[v], -100.0f) - lse_b);
                if (d != d) d = 0.0f;
                const float p = prev[i];
                const float q = qsr[i] + ALPHA_ * (p - llm) + BETA_ * d;
                qsr[i] = q;
                const float raw = (tf * LAM_ * llm + q + p * (1.0f / ETA_)) / denom;
                prev[i] = raw;
                mraw = fmaxf(mraw, raw);
            }
        }
        mraw = block_max(mraw, sred);
        float se = 0.0f;
        #pragma unroll
        for (int i = 0; i < ITEMS_C; ++i) {
            const int v = tid + i * THR_C;
            if (v < V_) se += expf(prev[i] - mraw);
        }
        se = block_sum(se, sred);
        const float lse = mraw + logf(se);
        #pragma unroll
        for (int i = 0; i < ITEMS_C; ++i) prev[i] -= lse;   // -inf stays -inf
    }

    // ---- write fused + argmax (first occurrence of max, like jnp.argmax) ----
    float bv = -INFINITY;
    int bi = 0x7fffffff;
    #pragma unroll
    for (int i = 0; i < ITEMS_C; ++i) {
        const int v = tid + i * THR_C;
        if (v < V_) {
            outrow[v] = prev[i];
            if (prev[i] > bv) { bv = prev[i]; bi = v; }
        }
    }
    #pragma unroll
    for (int o = 16; o > 0; o >>= 1) {
        const float vo = __shfl_xor(bv, o, 32);
        const int   io = __shfl_xor(bi, o, 32);
        if (vo > bv || (vo == bv && io < bi)) { bv = vo; bi = io; }
    }
    const int lane = tid & 31, w = tid >> 5;
    if (lane == 0) { sred[w] = bv; sredi[w] = bi; }
    __syncthreads();
    if (w == 0) {
        float x = (lane < (THR_C >> 5)) ? sred[lane] : -INFINITY;
        int   i = (lane < (THR_C >> 5)) ? sredi[lane] : 0x7fffffff;
        #pragma unroll
        for (int o = 16; o > 0; o >>= 1) {
            const float vo = __shfl_xor(x, o, 32);
            const int   io = __shfl_xor(i, o, 32);
            if (vo > x || (vo == x && io < i)) { x = vo; i = io; }
        }
        if (lane == 0) recovered[b] = i;
    }
}

// ---------------- kernel 4: emit output tokens ----------------
__global__ void emit_kernel(const int* __restrict__ draft,
                            const int* __restrict__ cnt,
                            const int* __restrict__ rejected,
                            const int* __restrict__ recovered,
                            int* __restrict__ out_tok) {
    const int b = threadIdx.x;
    if (b >= B_) return;
    const int c = cnt[b];
    const int rj = rejected[b];
    const int rec = recovered[b];
    #pragma unroll
    for (int k = 0; k < K_; ++k) {
        int v;
        if (k < c)               v = draft[b * K_ + k];
        else if (k == c && rj)   v = rec;
        else                     v = PLACEHOLDER_;
        out_tok[b * (K_ + 1) + k] = v;
    }
    out_tok[b * (K_ + 1) + K_] = PLACEHOLDER_;   // ENABLE_BONUS = False
}

// ---------------- launcher ----------------
extern "C" void kernel_launch(void* const* d_in, const int* in_sizes, int n_in,
                              void* d_out, int out_size, void* d_ws, size_t ws_size,
                              hipStream_t stream) {
    const float* target = (const float*)d_in[0];   // [B*K, V]
    const float* base   = (const float*)d_in[1];   // [B*K, V]
    const float* steer  = (const float*)d_in[2];   // [B*K, V]
    const int* draft    = (const int*)d_in[3];     // [B, K]
    const int* nd       = (const int*)d_in[4];     // [B]
    // d_in[5] = bonus_token_ids: unused (ENABLE_BONUS = False)

    // output layout (4-byte slots): tokens[B*(K+1)] int | cnt[B] int | fused[B*V] f32
    int*   out_tok = (int*)d_out;
    int*   out_cnt = out_tok + B_ * (K_ + 1);
    float* fused   = (float*)d_out + B_ * (K_ + 1) + B_;

    // workspace layout (~5 KB): p_llm | p_base | rp | rejected | recovered
    float* p_llm   = (float*)d_ws;
    float* p_base  = p_llm + B_ * K_;
    int*   rp      = (int*)(p_base + B_ * K_);
    int*   rejected  = rp + B_;
    int*   recovered = rejected + B_;

    probs_kernel<<<B_ * K_, THR_A, 0, stream>>>(target, base, draft, p_llm, p_base);
    accept_kernel<<<1, 64, 0, stream>>>(p_llm, p_base, nd, out_cnt, rp, rejected);
    costeer_kernel<<<B_, THR_C, 0, stream>>>(target, base, steer, rp, fused, recovered);
    emit_kernel<<<1, 64, 0, stream>>>(draft, out_cnt, rejected, recovered, out_tok);
}